// RoIBridge_67937792688165
// MI455X (gfx1250) — compile-verified
//
#include <hip/hip_runtime.h>

#define IMAGE_SIZE    224
#define POS_D         64
#define OBJ_TYPE_NUM  128
#define BATCH         2048

#define TABLE_ROWS    (IMAGE_SIZE + 1)        // 225
#define TABLE_FLOATS  (TABLE_ROWS * POS_D)    // 14400 floats = 57.6 KB
#define TABLE_CHUNKS  (TABLE_FLOATS / 4)      // 3600 float4 chunks

typedef float __attribute__((ext_vector_type(4))) f32x4;

#if __has_builtin(__builtin_amdgcn_global_load_async_to_lds_b128)
#define HAVE_ASYNC_LDS 1
#else
#define HAVE_ASYNC_LDS 0
#endif

#if HAVE_ASYNC_LDS
// The builtin's parameters are pointers to GCC-style <4 x i32> vectors in AS1/AS3.
typedef int v4i __attribute__((vector_size(16)));
typedef __attribute__((address_space(1))) v4i as1_v4i;
typedef __attribute__((address_space(3))) v4i as3_v4i;

// Generic->AS1: flat global addresses are numerically identical to AS1 addresses.
__device__ __forceinline__ as1_v4i* to_global_as1(const void* p) {
    return (as1_v4i*)(uintptr_t)p;
}
// Generic->AS3: generic shared addresses carry the LDS offset in the low 32 bits.
__device__ __forceinline__ as3_v4i* to_lds_as3(void* p) {
    return (as3_v4i*)(uint32_t)(uintptr_t)p;
}
#endif

__global__ void __launch_bounds__(256)
poe_gather_kernel(const float* __restrict__ bbox,   // [B, T, 4]
                  const int*   __restrict__ obj,    // [B, T]
                  const float* __restrict__ table,  // [225, 64]
                  float* __restrict__ out)          // [B*T, 256]
{
    __shared__ f32x4 lds_table[TABLE_CHUNKS];       // 57.6 KB per block

    const int tid = threadIdx.x;
    const f32x4* gtab = (const f32x4*)table;

    // ---- Stage the positional table into LDS (CDNA5 async global->LDS path) ----
#if HAVE_ASYNC_LDS
    for (int i = tid; i < TABLE_CHUNKS; i += 256) {
        __builtin_amdgcn_global_load_async_to_lds_b128(
            to_global_as1(gtab + i),
            to_lds_as3(&lds_table[i]),
            /*offset=*/0, /*cpol=*/0);
    }
#if __has_builtin(__builtin_amdgcn_s_wait_asynccnt)
    __builtin_amdgcn_s_wait_asynccnt(0);
#else
    asm volatile("s_wait_asynccnt 0" ::: "memory");
#endif
#else
    for (int i = tid; i < TABLE_CHUNKS; i += 256) {
        lds_table[i] = gtab[i];
    }
#endif
    __syncthreads();

    // ---- Streaming gather + mask: one float4 output chunk per thread-iteration ----
    // chunk i: row = i>>6 (256 floats/row = 64 chunks), coord = (i&63)>>4, d4 = i&15
    const int total_chunks = BATCH * OBJ_TYPE_NUM * 64;   // 16,777,216
    f32x4* out4 = (f32x4*)out;

    for (int i = blockIdx.x * 256 + tid; i < total_chunks;
         i += gridDim.x * 256) {
        const int row   = i >> 6;
        const int c     = i & 63;
        const int coord = c >> 4;
        const int d4    = c & 15;

        f32x4 v = {0.0f, 0.0f, 0.0f, 0.0f};
        if (obj[row] == 1) {
            float frac = bbox[(row << 2) + coord];
            float s = frac * (float)IMAGE_SIZE;
            s = fminf(fmaxf(s, 0.0f), (float)IMAGE_SIZE);
            const int idx = (int)s;                 // truncation, matches astype(int32)
            v = lds_table[idx * (POS_D / 4) + d4];  // conflict-free: consecutive banks
        }
        // Output is written once and never re-read: non-temporal B128 store.
        __builtin_nontemporal_store(v, out4 + i);
    }
}

extern "C" void kernel_launch(void* const* d_in, const int* in_sizes, int n_in,
                              void* d_out, int out_size, void* d_ws, size_t ws_size,
                              hipStream_t stream) {
    const float* bbox  = (const float*)d_in[0];  // [2048,128,4] f32
    const int*   obj   = (const int*)d_in[1];    // [2048,128]   i32
    const float* table = (const float*)d_in[2];  // [225,64]     f32
    float*       out   = (float*)d_out;          // [262144,256] f32

    (void)in_sizes; (void)n_in; (void)out_size; (void)d_ws; (void)ws_size;

    // 2048 blocks x 256 threads: each block amortizes its 57.6 KB table staging
    // over ~128 KB of output; grid-stride loop covers all 16.7M float4 chunks.
    poe_gather_kernel<<<dim3(2048), dim3(256), 0, stream>>>(bbox, obj, table, out);
}